// LIFSpike_78975858639622
// MI455X (gfx1250) — compile-verified
//
#include <hip/hip_runtime.h>

// LIF spike recurrence over T=8 (innermost, contiguous) time steps.
// Pure HBM-streaming problem: 256 MiB traffic, ~11.5 us floor @ 23.3 TB/s.
// CDNA5 path used: GLOBAL_LOAD_ASYNC_TO_LDS_B128 double-buffered prefetch
// (ASYNCcnt), split wait counters, non-temporal b128 stores.

#define LIF_TAU 0.25f
#define LIF_VTH 0.3f
#define TSTEPS  8
#define BLOCK   256
#define TILE_FLOATS (BLOCK * TSTEPS)   // 2048 floats = 8 KB per buffer

typedef float v4f __attribute__((ext_vector_type(4)));
typedef int   v4i __attribute__((ext_vector_type(4)));

#define AS1 __attribute__((address_space(1)))
#define AS3 __attribute__((address_space(3)))

#if defined(__has_builtin)
#  if __has_builtin(__builtin_amdgcn_global_load_async_to_lds_b128)
#    define USE_ASYNC 1
#  endif
#endif
#ifndef USE_ASYNC
#  define USE_ASYNC 0
#endif

#if USE_ASYNC
// Per-lane async copy of 16 bytes, global -> LDS.  Signature per hipcc
// diagnostic: (global int4*, shared int4*, imm offset, imm cpol).
__device__ __forceinline__ void async_copy_b128(const float* g, float* l) {
  __builtin_amdgcn_global_load_async_to_lds_b128(
      (AS1 v4i*)(void*)(const void*)g, (AS3 v4i*)l, 0, 0);
}
#endif

__device__ __forceinline__ void lif_step8(const float xi[TSTEPS], float ov[TSTEPS]) {
  float u = 0.0f, o = 0.0f;
#pragma unroll
  for (int t = 0; t < TSTEPS; ++t) {
    // u = TAU*u*(1-o) + x_t ; since o in {0,1}: u = o ? x_t : fma(TAU,u,x_t)
    float decayed = __builtin_fmaf(LIF_TAU, u, xi[t]);
    u = (o > 0.0f) ? xi[t] : decayed;
    o = (u > LIF_VTH) ? 1.0f : 0.0f;
    ov[t] = o;
  }
}

__global__ __launch_bounds__(BLOCK) void lif_kernel(const float* __restrict__ x,
                                                    float* __restrict__ out,
                                                    long long n_neurons) {
  const int tid = threadIdx.x;
  const long long nTiles = n_neurons / BLOCK;
  const long long stride = (long long)gridDim.x;
  long long cur = (long long)blockIdx.x;

#if USE_ASYNC
  __shared__ __align__(16) float smem[2][TILE_FLOATS];
  float* l0 = &smem[0][tid * TSTEPS];
  float* l1 = &smem[1][tid * TSTEPS];

  if (cur < nTiles) {  // uniform branch: EXEC stays all-ones for async ops
    {
      const float* g = x + cur * (long long)TILE_FLOATS + (long long)(tid * TSTEPS);
      async_copy_b128(g, l0);
      async_copy_b128(g + 4, l0 + 4);
    }
    int buf = 0;
    while (cur < nTiles) {
      long long nxt = cur + stride;
      float* lc = buf ? l1 : l0;
      float* ln = buf ? l0 : l1;
      if (nxt < nTiles) {
        const float* g = x + nxt * (long long)TILE_FLOATS + (long long)(tid * TSTEPS);
        async_copy_b128(g, ln);
        async_copy_b128(g + 4, ln + 4);
        asm volatile("s_wait_asynccnt 2" ::: "memory");  // current tile's 2 copies done
      } else {
        asm volatile("s_wait_asynccnt 0" ::: "memory");
      }
      v4f a = *(const v4f*)lc;
      v4f b = *(const v4f*)(lc + 4);
      float xi[TSTEPS] = {a.x, a.y, a.z, a.w, b.x, b.y, b.z, b.w};
      float ov[TSTEPS];
      lif_step8(xi, ov);
      v4f oa = {ov[0], ov[1], ov[2], ov[3]};
      v4f ob = {ov[4], ov[5], ov[6], ov[7]};
      float* dst = out + cur * (long long)TILE_FLOATS + (long long)(tid * TSTEPS);
      __builtin_nontemporal_store(oa, (v4f*)dst);
      __builtin_nontemporal_store(ob, (v4f*)(dst + 4));
      // WAR guard: DS reads of this buffer must land before the async engine
      // overwrites it next iteration (async LDS writes are unordered vs DS).
      asm volatile("s_wait_dscnt 0" ::: "memory");
      cur = nxt;
      buf ^= 1;
    }
  }
#else
  for (; cur < nTiles; cur += stride) {
    const float* src = x + cur * (long long)TILE_FLOATS + (long long)(tid * TSTEPS);
    v4f a = __builtin_nontemporal_load((const v4f*)src);
    v4f b = __builtin_nontemporal_load((const v4f*)(src + 4));
    float xi[TSTEPS] = {a.x, a.y, a.z, a.w, b.x, b.y, b.z, b.w};
    float ov[TSTEPS];
    lif_step8(xi, ov);
    v4f oa = {ov[0], ov[1], ov[2], ov[3]};
    v4f ob = {ov[4], ov[5], ov[6], ov[7]};
    float* dst = out + cur * (long long)TILE_FLOATS + (long long)(tid * TSTEPS);
    __builtin_nontemporal_store(oa, (v4f*)dst);
    __builtin_nontemporal_store(ob, (v4f*)(dst + 4));
  }
#endif

  // Tail: n_neurons % 256 leftover neurons (zero for the reference shape).
  long long tailBase = nTiles * (long long)BLOCK;
  if (blockIdx.x == 0) {
    long long idx = tailBase + (long long)tid;
    if (idx < n_neurons) {
      const float* src = x + idx * TSTEPS;
      float xi[TSTEPS], ov[TSTEPS];
#pragma unroll
      for (int t = 0; t < TSTEPS; ++t) xi[t] = src[t];
      lif_step8(xi, ov);
      float* dst = out + idx * TSTEPS;
#pragma unroll
      for (int t = 0; t < TSTEPS; ++t) dst[t] = ov[t];
    }
  }
}

extern "C" void kernel_launch(void* const* d_in, const int* in_sizes, int n_in,
                              void* d_out, int out_size, void* d_ws, size_t ws_size,
                              hipStream_t stream) {
  (void)n_in; (void)out_size; (void)d_ws; (void)ws_size;
  const float* x = (const float*)d_in[0];
  float* out = (float*)d_out;
  long long n_elem = (long long)in_sizes[0];        // 33,554,432
  long long n_neurons = n_elem / TSTEPS;            // 4,194,304
  long long n_tiles = n_neurons / BLOCK;            // 16,384
  long long grid = n_tiles < 2048 ? (n_tiles > 0 ? n_tiles : 1) : 2048;
  lif_kernel<<<(int)grid, BLOCK, 0, stream>>>(x, out, n_neurons);
}